// SHOTDescriptor_64622077935970
// MI455X (gfx1250) — compile-verified
//
#include <hip/hip_runtime.h>
#include <math.h>

typedef __attribute__((ext_vector_type(2))) float v2f;
typedef __attribute__((ext_vector_type(8))) float v8f;

#define EPS_REF 1e-5f

constexpr int TILE    = 16;
constexpr int COLS    = 128;   // candidate columns staged in LDS per chunk (16384 % 128 == 0)
constexpr int RSTRIDE = 130;   // dword row stride: bank = (2*row + col) % 64 -> conflict-free
constexpr int WAVES   = 2;     // waves per block -> 33 KB static LDS

// ---------------------------------------------------------------------------
// Prep: A4[j] = (x,y,z,1)  (WMMA A rows, K=4)
//       Qv[j] = (-x,-y,-z,|p|^2/2)  (WMMA B cols) so C[m][n] = (d2 - |pm|^2)/2,
//       which is order-equivalent to d2 within each query row.
// ---------------------------------------------------------------------------
__global__ void prep_kernel(const float* __restrict__ P,
                            float4* __restrict__ A4,
                            float4* __restrict__ Qv, int n) {
  int j = blockIdx.x * blockDim.x + threadIdx.x;
  if (j >= n) return;
  float x = P[3 * j + 0], y = P[3 * j + 1], z = P[3 * j + 2];
  A4[j] = make_float4(x, y, z, 1.0f);
  Qv[j] = make_float4(-x, -y, -z, 0.5f * (x * x + y * y + z * z));
}

// Branchless sorted-insert into ascending 5-list (executed only under early-out).
__device__ __forceinline__ void insert5(float s, int j, float d[5], int id[5]) {
  float x = s; int xi = j;
#pragma unroll
  for (int i = 0; i < 5; ++i) {
    bool c = x < d[i];
    float nd = c ? x : d[i];
    float od = c ? d[i] : x;
    int   ni = c ? xi : id[i];
    int   oi = c ? id[i] : xi;
    d[i] = nd; id[i] = ni; x = od; xi = oi;
  }
}

// Eigenvector of symmetric 3x3 for eigenvalue lam: largest cross product of
// rows of (A - lam*I), with fallback axis for degenerate cases.
__device__ __forceinline__ void sym3_eigvec(
    float a00, float a01, float a02, float a11, float a12, float a22,
    float lam, float fx, float fy, float fz,
    float* vx, float* vy, float* vz) {
  float r0x = a00 - lam, r0y = a01,       r0z = a02;
  float r1x = a01,       r1y = a11 - lam, r1z = a12;
  float r2x = a02,       r2y = a12,       r2z = a22 - lam;
  float cax = r0y*r1z - r0z*r1y, cay = r0z*r1x - r0x*r1z, caz = r0x*r1y - r0y*r1x;
  float cbx = r0y*r2z - r0z*r2y, cby = r0z*r2x - r0x*r2z, cbz = r0x*r2y - r0y*r2x;
  float ccx = r1y*r2z - r1z*r2y, ccy = r1z*r2x - r1x*r2z, ccz = r1x*r2y - r1y*r2x;
  float na = cax*cax + cay*cay + caz*caz;
  float nb = cbx*cbx + cby*cby + cbz*cbz;
  float nc = ccx*ccx + ccy*ccy + ccz*ccz;
  float bx = cax, by = cay, bz = caz, bn = na;
  if (nb > bn) { bx = cbx; by = cby; bz = cbz; bn = nb; }
  if (nc > bn) { bx = ccx; by = ccy; bz = ccz; bn = nc; }
  if (bn > 1e-30f) {
    float s = 1.0f / sqrtf(bn);
    *vx = bx * s; *vy = by * s; *vz = bz * s;
  } else {
    *vx = fx; *vy = fy; *vz = fz;
  }
}

// ---------------------------------------------------------------------------
// Main: one wave per 32-query strip. Per 128-candidate chunk:
//   produce: 8 col-tiles x 2 row-tiles of V_WMMA_F32_16X16X4_F32 -> LDS scores
//   scan:    lane L owns query q0+L; early-exit top-5 over its LDS row
// Then per-lane covariance / eigendecomposition / sign votes / output.
// ---------------------------------------------------------------------------
__global__ __launch_bounds__(WAVES * 32)
void knn_lrf_kernel(const float4* __restrict__ A4,
                    const float4* __restrict__ Qv,
                    float* __restrict__ out, int n) {
  __shared__ float lds[WAVES * 32 * RSTRIDE];
  const int wave = threadIdx.x >> 5;
  const int lane = threadIdx.x & 31;
  const int q0 = (blockIdx.x * WAVES + wave) * 32;
  if (q0 >= n) return;

  float* buf = &lds[wave * 32 * RSTRIDE];
  const int half = lane >> 4;   // 0: K0,K1 half  1: K2,K3 half
  const int l16  = lane & 15;

  const float* A4f = reinterpret_cast<const float*>(A4);
  const float* Qf  = reinterpret_cast<const float*>(Qv);

  // A fragments: lane L<16 holds {x,y} (K0,K1) of query m=L; L>=16 holds {z,1} (K2,K3).
  v2f a0 = *reinterpret_cast<const v2f*>(A4f + (size_t)(q0 + l16) * 4 + half * 2);
  v2f a1 = *reinterpret_cast<const v2f*>(A4f + (size_t)(q0 + 16 + l16) * 4 + half * 2);

  float d5[5]; int id5[5];
#pragma unroll
  for (int i = 0; i < 5; ++i) { d5[i] = 3.0e38f; id5[i] = 0; }

  const float* rowp = buf + lane * RSTRIDE;

  for (int cb = 0; cb < n; cb += COLS) {
    // ---- produce score tiles via WMMA ----
#pragma unroll
    for (int t = 0; t < COLS / TILE; ++t) {
      int j0 = cb + t * TILE;
      // B fragment: lane L<16 holds {-x,-y} (K0,K1) of candidate n=L; L>=16 holds {-z,|p|^2/2}.
      v2f b = *reinterpret_cast<const v2f*>(Qf + (size_t)(j0 + l16) * 4 + half * 2);
      v8f cz = {};
      v8f c0 = __builtin_amdgcn_wmma_f32_16x16x4_f32(false, a0, false, b,
                                                     (short)0, cz, false, false);
      v8f c1 = __builtin_amdgcn_wmma_f32_16x16x4_f32(false, a1, false, b,
                                                     (short)0, cz, false, false);
      // C layout: lanes 0-15 -> rows r (VGPR r), col = lane; lanes 16-31 -> rows r+8.
      int col   = t * TILE + l16;
      float* p0 = buf + (half * 8) * RSTRIDE + col;
      float* p1 = buf + (16 + half * 8) * RSTRIDE + col;
#pragma unroll
      for (int r = 0; r < 8; ++r) {
        p0[r * RSTRIDE] = c0[r];
        p1[r * RSTRIDE] = c1[r];
      }
    }
    asm volatile("" ::: "memory");  // keep produce (DS stores) before scan (DS loads)

    // ---- scan own row with early-exit top-5 ----
#pragma unroll 4
    for (int c = 0; c < COLS; c += 2) {
      float s0 = rowp[c];
      float s1 = rowp[c + 1];
      if (fminf(s0, s1) < d5[4]) {       // ~90% of iterations skip via s_cbranch_execz
        insert5(s0, cb + c,     d5, id5);
        insert5(s1, cb + c + 1, d5, id5);
      }
    }
    asm volatile("" ::: "memory");  // keep scan before next chunk overwrites buf
  }

  // ---------------- per-point LRF (lane = query) ----------------
  const int q = q0 + lane;
  float4 pq4 = A4[q];
  float px = pq4.x, py = pq4.y, pz = pq4.z;

  float nbx[5], nby[5], nbz[5];
#pragma unroll
  for (int k = 0; k < 5; ++k) {
    float4 nb4 = A4[id5[k]];
    nbx[k] = nb4.x; nby[k] = nb4.y; nbz[k] = nb4.z;
  }

  // Centered offsets: (nb - p) - mean(nb - p) = nb - mean(nb); p cancels.
  float sx = 0.f, sy = 0.f, sz = 0.f;
#pragma unroll
  for (int k = 0; k < 5; ++k) { sx += nbx[k]; sy += nby[k]; sz += nbz[k]; }
  float mx = 0.2f * sx, my = 0.2f * sy, mz = 0.2f * sz;

  float cxx = 0.f, cxy = 0.f, cxz = 0.f, cyy = 0.f, cyz = 0.f, czz = 0.f;
#pragma unroll
  for (int k = 0; k < 5; ++k) {
    float ex = nbx[k] - mx, ey = nby[k] - my, ez = nbz[k] - mz;
    cxx += ex * ex; cxy += ex * ey; cxz += ex * ez;
    cyy += ey * ey; cyz += ey * ez; czz += ez * ez;
  }
  cxx *= 0.2f; cxy *= 0.2f; cxz *= 0.2f; cyy *= 0.2f; cyz *= 0.2f; czz *= 0.2f;

  // Closed-form symmetric 3x3 eigendecomposition (ascending e0<=e1<=e2).
  float v0x, v0y, v0z, v1x, v1y, v1z, v2x, v2y, v2z;
  float q3 = (cxx + cyy + czz) * (1.0f / 3.0f);
  float b00 = cxx - q3, b11 = cyy - q3, b22 = czz - q3;
  float p1 = cxy * cxy + cxz * cxz + cyz * cyz;
  float p2 = b00 * b00 + b11 * b11 + b22 * b22 + 2.0f * p1;
  if (p2 < 1e-24f) {
    v0x = 1.f; v0y = 0.f; v0z = 0.f;
    v1x = 0.f; v1y = 1.f; v1z = 0.f;
    v2x = 0.f; v2y = 0.f; v2z = 1.f;
  } else {
    float pp  = sqrtf(p2 * (1.0f / 6.0f));
    float inv = 1.0f / pp;
    float c00 = b00 * inv, c11 = b11 * inv, c22 = b22 * inv;
    float c01 = cxy * inv, c02 = cxz * inv, c12 = cyz * inv;
    float detB = c00 * (c11 * c22 - c12 * c12)
               - c01 * (c01 * c22 - c12 * c02)
               + c02 * (c01 * c12 - c11 * c02);
    float r   = fminf(1.0f, fmaxf(-1.0f, 0.5f * detB));
    float phi = acosf(r) * (1.0f / 3.0f);
    float e2  = q3 + 2.0f * pp * cosf(phi);
    float e0  = q3 + 2.0f * pp * cosf(phi + 2.0943951023931953f);
    sym3_eigvec(cxx, cxy, cxz, cyy, cyz, czz, e2, 0.f, 0.f, 1.f, &v2x, &v2y, &v2z);
    sym3_eigvec(cxx, cxy, cxz, cyy, cyz, czz, e0, 1.f, 0.f, 0.f, &v0x, &v0y, &v0z);
    v1x = v2y * v0z - v2z * v0y;
    v1y = v2z * v0x - v2x * v0z;
    v1z = v2x * v0y - v2y * v0x;
    float n1 = v1x * v1x + v1y * v1y + v1z * v1z;
    if (n1 > 1e-30f) {
      float s = 1.0f / sqrtf(n1);
      v1x *= s; v1y *= s; v1z *= s;
    } else { v1x = 0.f; v1y = 1.f; v1z = 0.f; }
  }

  // Faithful quirk: axes are ROWS of eigvecs; order is always [2,1,0].
  float xrx = v0z, xry = v1z, xrz = v2z;   // row 2 -> x-axis candidate
  float zrx = v0x, zry = v1x, zrz = v2x;   // row 0 -> z-axis candidate

  // Faithful quirk: vote vector = diffs - p = nb - 2p.
  int cx_cnt = 0, cz_cnt = 0;
#pragma unroll
  for (int k = 0; k < 5; ++k) {
    float vx = nbx[k] - 2.0f * px;
    float vy = nby[k] - 2.0f * py;
    float vz = nbz[k] - 2.0f * pz;
    cx_cnt += (vx * xrx + vy * xry + vz * xrz >= -EPS_REF) ? 1 : 0;
    cz_cnt += (vx * zrx + vy * zry + vz * zrz >= -EPS_REF) ? 1 : 0;
  }
  float sxn = (2 * cx_cnt >= 5) ? 1.0f : -1.0f;
  float szn = (2 * cz_cnt >= 5) ? 1.0f : -1.0f;
  float X0 = xrx * sxn, X1 = xry * sxn, X2 = xrz * sxn;
  float Z0 = zrx * szn, Z1 = zry * szn, Z2 = zrz * szn;
  float Y0 = Z1 * X2 - Z2 * X1;
  float Y1 = Z2 * X0 - Z0 * X2;
  float Y2 = Z0 * X1 - Z1 * X0;

  float* o = out + (size_t)q * 9;
  o[0] = X0; o[1] = X1; o[2] = X2;
  o[3] = Y0; o[4] = Y1; o[5] = Y2;
  o[6] = Z0; o[7] = Z1; o[8] = Z2;
}

extern "C" void kernel_launch(void* const* d_in, const int* in_sizes, int n_in,
                              void* d_out, int out_size, void* d_ws, size_t ws_size,
                              hipStream_t stream) {
  (void)n_in; (void)out_size; (void)ws_size;
  const float* P = (const float*)d_in[0];
  float* out = (float*)d_out;
  const int n = in_sizes[0] / 3;       // 16384

  // Workspace: A4 (n*16 B) + Qv (n*16 B) = 512 KB.
  float4* A4 = reinterpret_cast<float4*>(d_ws);
  float4* Qv = A4 + n;

  prep_kernel<<<(n + 255) / 256, 256, 0, stream>>>(P, A4, Qv, n);

  const int strips = n / 32;                       // 512 waves of strip work
  const int blocks = (strips + WAVES - 1) / WAVES; // 256 blocks of 2 waves
  knn_lrf_kernel<<<blocks, WAVES * 32, 0, stream>>>(A4, Qv, out, n);
}